// YOLOLoss_57312043598252
// MI455X (gfx1250) — compile-verified
//
#include <hip/hip_runtime.h>
#include <math.h>

// ---------------------------------------------------------------------------
// YOLO loss for MI455X (gfx1250, wave32).
// Strategy: compact sparse targets (obj != 0) per image, then best-IoU over
// the short compacted list instead of the full P x P pairwise matrix.
// Deterministic reduction: per-block shared-mem tree -> 99 partials ->
// single-wave V_WMMA_F32_16X16X4_F32 ones-matmul reduction.
// ---------------------------------------------------------------------------

typedef __attribute__((ext_vector_type(2))) float v2f;
typedef __attribute__((ext_vector_type(8))) float v8f;

__device__ __forceinline__ float bce_logits(float label, float logit) {
    // max(x,0) - x*z + log1p(exp(-|x|))
    return fmaxf(logit, 0.0f) - logit * label + log1pf(expf(-fabsf(logit)));
}

__device__ __forceinline__ float sigmoidf(float x) {
    return 1.0f / (1.0f + expf(-x));
}

__global__ void zero_counters_kernel(int* cnt) {
    if (threadIdx.x < 12) cnt[threadIdx.x] = 0;
}

// One thread per (b, row, col, anchor). Append valid target boxes to per-image list.
__global__ void compact_targets_kernel(const float* __restrict__ t, int P, int total,
                                       float* __restrict__ boxes, int* __restrict__ cnt) {
    int idx = blockIdx.x * blockDim.x + threadIdx.x;
    if (idx >= total) return;
    const float* tp = t + (size_t)idx * 85;
    float obj = tp[4];
    if (obj != 0.0f) {
        int b = idx / P;
        int pos = atomicAdd(&cnt[b], 1);
        float4 v;
        v.x = tp[0]; v.y = tp[1]; v.z = tp[2]; v.w = tp[3];
        ((float4*)boxes)[(size_t)b * P + pos] = v;
    }
}

__global__ void __launch_bounds__(256) yolo_layer_loss_kernel(
    const float* __restrict__ t, const float* __restrict__ y,
    int g, int P, int total,
    float ax0, float ay0, float ax1, float ay1, float ax2, float ay2,
    const float* __restrict__ tboxes, const int* __restrict__ cnt,
    float* __restrict__ partials, int partial_off)
{
    int idx = blockIdx.x * blockDim.x + threadIdx.x;
    float sum = 0.0f;
    if (idx < total) {
        int b    = idx / P;
        int r    = idx - b * P;
        int a    = r % 3;
        int cell = r / 3;
        int col  = cell % g;   // grid x
        int row  = cell / g;   // grid y
        const float* tp = t + (size_t)idx * 85;
        const float* yp = y + (size_t)idx * 85;

        float t0 = tp[0], t1 = tp[1], t2 = tp[2], t3 = tp[3], obj = tp[4];
        float y0 = yp[0], y1 = yp[1], y2 = yp[2], y3 = yp[3], y4 = yp[4];

        float aw = (a == 0) ? ax0 : ((a == 1) ? ax1 : ax2);
        float ah = (a == 0) ? ay0 : ((a == 1) ? ay1 : ay2);

        float gf = (float)g;
        // predicted box (yolo head)
        float px = (sigmoidf(y0) + (float)col) / gf;
        float py = (sigmoidf(y1) + (float)row) / gf;
        float pw = expf(y2) * aw * (1.0f / 320.0f);
        float ph = expf(y3) * ah * (1.0f / 320.0f);
        float p_area = pw * ph;
        float pxmin = px - pw * 0.5f, pxmax = px + pw * 0.5f;
        float pymin = py - ph * 0.5f, pymax = py + ph * 0.5f;

        // best IoU against compacted valid targets of this image
        int n = cnt[b];
        const float4* tb = (const float4*)tboxes + (size_t)b * P;
        float best = -__builtin_huge_valf();
        for (int j = 0; j < n; ++j) {
            float4 q = tb[j];
            float txmin = q.x - q.z * 0.5f, txmax = q.x + q.z * 0.5f;
            float tymin = q.y - q.w * 0.5f, tymax = q.y + q.w * 0.5f;
            float wx = fmaxf(fminf(pxmax, txmax) - fmaxf(pxmin, txmin), 0.0f);
            float wy = fmaxf(fminf(pymax, tymax) - fmaxf(pymin, tymin), 0.0f);
            float inter = wx * wy;
            float uni = p_area + q.z * q.w - inter;
            best = fmaxf(best, inter / uni);
        }
        float ignore = (best < 0.5f) ? 1.0f : 0.0f;

        // confidence loss (all cells)
        float w_conf = obj + (1.0f - obj) * ignore;
        float loss = w_conf * bce_logits(obj, y4);

        if (obj != 0.0f) {
            float box_scale = 2.0f - t2 * t3;
            float rtx = t0 * gf - (float)col;
            float rty = t1 * gf - (float)row;
            loss += obj * box_scale * (bce_logits(rtx, y0) + bce_logits(rty, y1));
            float rtw = logf(t2 * 320.0f / aw);
            float rth = logf(t3 * 320.0f / ah);
            float dw = rtw - y2, dh = rth - y3;
            loss += obj * box_scale * 0.5f * (dw * dw + dh * dh);
            float cls = 0.0f;
            for (int k = 0; k < 80; ++k)
                cls += bce_logits(tp[5 + k], yp[5 + k]);
            loss += obj * cls;
        }
        sum = loss * 0.25f;  // mean over batch (B=4)
    }

    // deterministic block tree reduction
    __shared__ float sdata[256];
    sdata[threadIdx.x] = sum;
    __syncthreads();
    for (int s = 128; s > 0; s >>= 1) {
        if (threadIdx.x < s) sdata[threadIdx.x] += sdata[threadIdx.x + s];
        __syncthreads();
    }
    if (threadIdx.x == 0) partials[partial_off + blockIdx.x] = sdata[0];
}

// Single-wave final reduction through the matrix pipe:
// D(16x16) = Ones(16x4) x B(4x16) + C  => every D slot accumulates a column sum.
// Sum all 256 C slots (layout-agnostic) and divide by 16 (exact).
__global__ void final_reduce_wmma_kernel(const float* __restrict__ partials, int n,
                                         float* __restrict__ out) {
    int lane = threadIdx.x;           // launched with exactly 32 threads: EXEC all-ones
    v2f a; a[0] = 1.0f; a[1] = 1.0f;  // all-ones A matrix (16x4 f32 = 2 VGPRs/lane)
    v8f c = {};
    for (int base = 0; base < n; base += 64) {
        int i0 = base + lane;
        int i1 = base + 32 + lane;
        v2f bv;
        bv[0] = (i0 < n) ? partials[i0] : 0.0f;
        bv[1] = (i1 < n) ? partials[i1] : 0.0f;
        c = __builtin_amdgcn_wmma_f32_16x16x4_f32(
                /*neg_a=*/false, a, /*neg_b=*/false, bv,
                /*c_mod=*/(short)0, c, /*reuse_a=*/false, /*reuse_b=*/false);
    }
    float v = c[0] + c[1] + c[2] + c[3] + c[4] + c[5] + c[6] + c[7];
    for (int m = 16; m > 0; m >>= 1) v += __shfl_xor(v, m, 32);
    if (lane == 0) out[0] = v * (1.0f / 16.0f);
}

extern "C" void kernel_launch(void* const* d_in, const int* in_sizes, int n_in,
                              void* d_out, int out_size, void* d_ws, size_t ws_size,
                              hipStream_t stream) {
    // setup_inputs() dict order: t0, y0, t1, y1, t2, y2
    const float* t0 = (const float*)d_in[0];
    const float* y0 = (const float*)d_in[1];
    const float* t1 = (const float*)d_in[2];
    const float* y1 = (const float*)d_in[3];
    const float* t2 = (const float*)d_in[4];
    const float* y2 = (const float*)d_in[5];

    const int B = 4;
    const int P0 = 10 * 10 * 3, P1 = 20 * 20 * 3, P2 = 40 * 40 * 3;  // 300,1200,4800
    const int N0 = B * P0, N1 = B * P1, N2 = B * P2;                 // 1200,4800,19200

    float* ws  = (float*)d_ws;
    int*   cnt = (int*)d_ws;           // 12 counters (layer*4 + b), bytes [0,48)
    float* tb0 = ws + 64;              // B*P0*4 = 4800 floats
    float* tb1 = ws + 64 + 4800;       // B*P1*4 = 19200 floats
    float* tb2 = ws + 64 + 4800 + 19200;       // B*P2*4 = 76800 floats
    float* partials = ws + 64 + 4800 + 19200 + 76800;

    const int BLK = 256;
    int nb0 = (N0 + BLK - 1) / BLK;    // 5
    int nb1 = (N1 + BLK - 1) / BLK;    // 19
    int nb2 = (N2 + BLK - 1) / BLK;    // 75
    int npart = nb0 + nb1 + nb2;       // 99

    zero_counters_kernel<<<1, 32, 0, stream>>>(cnt);

    compact_targets_kernel<<<nb0, BLK, 0, stream>>>(t0, P0, N0, tb0, cnt + 0);
    compact_targets_kernel<<<nb1, BLK, 0, stream>>>(t1, P1, N1, tb1, cnt + 4);
    compact_targets_kernel<<<nb2, BLK, 0, stream>>>(t2, P2, N2, tb2, cnt + 8);

    // Anchors: layer0 (g=10) mask [6,7,8]; layer1 (g=20) mask [3,4,5]; layer2 (g=40) mask [0,1,2]
    yolo_layer_loss_kernel<<<nb0, BLK, 0, stream>>>(
        t0, y0, 10, P0, N0,
        116.0f, 90.0f, 156.0f, 198.0f, 373.0f, 326.0f,
        tb0, cnt + 0, partials, 0);
    yolo_layer_loss_kernel<<<nb1, BLK, 0, stream>>>(
        t1, y1, 20, P1, N1,
        30.0f, 61.0f, 62.0f, 45.0f, 59.0f, 119.0f,
        tb1, cnt + 4, partials, nb0);
    yolo_layer_loss_kernel<<<nb2, BLK, 0, stream>>>(
        t2, y2, 40, P2, N2,
        10.0f, 13.0f, 16.0f, 30.0f, 33.0f, 23.0f,
        tb2, cnt + 8, partials, nb0 + nb1);

    final_reduce_wmma_kernel<<<1, 32, 0, stream>>>(partials, npart, (float*)d_out);
}